// SelfAttention_1597727834253
// MI455X (gfx1250) — compile-verified
//
#include <hip/hip_runtime.h>

// ---------------------------------------------------------------------------
// SelfAttention (Chebyshev graph-conv QKV + head-mixing grouped softmax) for
// MI455X / gfx1250.  All GEMMs run through V_WMMA_F32_16X16X4_F32 (exact fp32
// matrix path).  lap (67MB) fits in the 192MB L2, so the problem is
// matrix-ALU bound, not HBM bound.  GEMM global->LDS staging uses CDNA5
// GLOBAL_LOAD_ASYNC_TO_LDS_B128 with double-buffered LDS tiles (ASYNCcnt).
// ---------------------------------------------------------------------------

typedef float v2f __attribute__((ext_vector_type(2)));
typedef float v8f __attribute__((ext_vector_type(8)));

#define B_    2
#define L_    4096
#define C_    256
#define H_    4
#define HD_   64
#define KCH_  4            // Chebyshev order
#define KC_   (KCH_ * C_)  // 1024 stacked features
#define NG_   32           // GroupNorm groups
#define EPS_  1e-5f

__device__ __forceinline__ v8f vzero8() {
  v8f z = {0.f,0.f,0.f,0.f,0.f,0.f,0.f,0.f};
  return z;
}

// D = A(16x4) * B(4x16) + C(16x16), all fp32.
__device__ __forceinline__ v8f wmma4(v2f a, v2f b, v8f c) {
  return __builtin_amdgcn_wmma_f32_16x16x4_f32(
      /*neg_a=*/false, a, /*neg_b=*/false, b,
      /*c_mod=*/(short)0, c, /*reuse_a=*/false, /*reuse_b=*/false);
}

// Async copy of 16B/lane from global to LDS (ASYNCcnt-tracked).
// ldsaddr: LDS byte offset (low 32 bits of flat-shared pointer, ISA 10.2).
__device__ __forceinline__ void async_ld_b128(unsigned ldsaddr, const float* g) {
  asm volatile("global_load_async_to_lds_b128 %0, %1, off"
               :: "v"(ldsaddr), "v"(g) : "memory");
}
__device__ __forceinline__ unsigned lds_off(const void* p) {
  return (unsigned)(unsigned long long)p;
}

// ---------------------------------------------------------------------------
// GroupNorm: stats per (batch, group) over (L, C/G) = 4096*8 elements
// ---------------------------------------------------------------------------
__global__ __launch_bounds__(256) void gn_stats_kernel(
    const float* __restrict__ x, float* __restrict__ stats) {
  const int bg = blockIdx.x;            // [0, B_*NG_)
  const int b  = bg >> 5;
  const int g  = bg & 31;
  __shared__ float s_sum[256];
  __shared__ float s_sq[256];
  float s = 0.f, sq = 0.f;
  for (int idx = threadIdx.x; idx < L_ * 8; idx += 256) {
    const int l = idx >> 3, j = idx & 7;
    const float v = x[((long)(b * L_ + l)) * C_ + g * 8 + j];
    s += v; sq += v * v;
  }
  s_sum[threadIdx.x] = s;
  s_sq[threadIdx.x]  = sq;
  __syncthreads();
  for (int st = 128; st > 0; st >>= 1) {
    if ((int)threadIdx.x < st) {
      s_sum[threadIdx.x] += s_sum[threadIdx.x + st];
      s_sq[threadIdx.x]  += s_sq[threadIdx.x + st];
    }
    __syncthreads();
  }
  if (threadIdx.x == 0) {
    const float inv_n = 1.f / (float)(L_ * 8);
    const float mean = s_sum[0] * inv_n;
    const float var  = s_sq[0] * inv_n - mean * mean;
    stats[bg * 2 + 0] = mean;
    stats[bg * 2 + 1] = rsqrtf(var + EPS_);
  }
}

// Apply norm and write directly into T0 slice of tcat [B, L, KC_]
__global__ __launch_bounds__(256) void gn_apply_kernel(
    const float* __restrict__ x, const float* __restrict__ gamma,
    const float* __restrict__ beta, const float* __restrict__ stats,
    float* __restrict__ tcat) {
  const long i = (long)blockIdx.x * 256 + threadIdx.x;   // over B*L*C
  const int  c  = (int)(i & (C_ - 1));
  const long bl = i >> 8;
  const int  b  = (int)(bl >> 12);
  const int  g  = c >> 3;
  const float mean = stats[(b * NG_ + g) * 2 + 0];
  const float rstd = stats[(b * NG_ + g) * 2 + 1];
  tcat[bl * KC_ + c] = (x[i] - mean) * rstd * gamma[c] + beta[c];
}

// ---------------------------------------------------------------------------
// Generic fp32 WMMA GEMM:  D = alpha * A[M,K] @ B[K,N] + beta * Z
// Block tile 128x64, 256 threads = 8 waves (4 along M x 2 along N),
// each wave owns a 32x32 tile = 2x2 WMMA fragments.
// Global->LDS staging via async-to-LDS copies, double-buffered (ASYNCcnt).
// ---------------------------------------------------------------------------
__global__ __launch_bounds__(256) void gemm_wmma_kernel(
    const float* __restrict__ A, int lda, long sA,
    const float* __restrict__ Bm, int ldb, long sB,
    float* __restrict__ Cm, int ldc, long sC,
    const float* __restrict__ Z, int ldz, long sZ,
    int Kd, float alpha, float beta) {
  __shared__ float As[2][128][16];
  __shared__ float Bs[2][16][64];

  const int tid  = threadIdx.x;
  const int lane = tid & 31;
  const int w    = tid >> 5;
  const int wm   = (w >> 1) * 32;        // wave M offset in block tile
  const int wn   = (w & 1) * 32;         // wave N offset in block tile
  const int mblock = blockIdx.y * 128;
  const int nblock = blockIdx.x * 64;

  const float* Ab = A  + (long)blockIdx.z * sA;
  const float* Bb = Bm + (long)blockIdx.z * sB;
  float*       Cb = Cm + (long)blockIdx.z * sC;

  v8f acc[2][2];
  acc[0][0] = vzero8(); acc[0][1] = vzero8();
  acc[1][0] = vzero8(); acc[1][1] = vzero8();

  const int rA = tid >> 2, cA = (tid & 3) * 4;    // A tile: 128x16 (2 rows/thread)
  const int rB = tid >> 4, cB = (tid & 15) * 4;   // B tile: 16x64

  // Per-thread global base pointers (advance by kb along K)
  const float* gA0 = Ab + (long)(mblock + rA) * lda + cA;
  const float* gA1 = Ab + (long)(mblock + rA + 64) * lda + cA;
  const float* gB  = Bb + (long)rB * ldb + nblock + cB;

  // Per-thread LDS destinations for both buffers
  unsigned dA0[2], dA1[2], dB[2];
#pragma unroll
  for (int p = 0; p < 2; p++) {
    dA0[p] = lds_off(&As[p][rA][cA]);
    dA1[p] = lds_off(&As[p][rA + 64][cA]);
    dB[p]  = lds_off(&Bs[p][rB][cB]);
  }

  // Prologue: stream tile kb=0 into buffer 0
  async_ld_b128(dA0[0], gA0);
  async_ld_b128(dA1[0], gA1);
  async_ld_b128(dB[0],  gB);

  for (int kb = 0; kb < Kd; kb += 16) {
    const int  cur  = (kb >> 4) & 1;
    const bool more = (kb + 16) < Kd;
    if (more) {   // stream next tile into the other buffer (read-done last iter)
      const int nxt = cur ^ 1;
      async_ld_b128(dA0[nxt], gA0 + (kb + 16));
      async_ld_b128(dA1[nxt], gA1 + (kb + 16));
      async_ld_b128(dB[nxt],  gB + (long)(kb + 16) * ldb);
      asm volatile("s_wait_asynccnt 0x3" ::: "memory");  // own cur-tile copies done
    } else {
      asm volatile("s_wait_asynccnt 0x0" ::: "memory");
    }
    __syncthreads();  // all waves' cur-tile copies visible

#pragma unroll
    for (int kk = 0; kk < 4; kk++) {
      const int c2 = kk * 4 + ((lane >> 4) << 1);  // K pair per lane half
      const int ar = wm + (lane & 15);
      v2f a0 = *(v2f*)&As[cur][ar][c2];
      v2f a1 = *(v2f*)&As[cur][ar + 16][c2];
      const int nc = wn + (lane & 15);
      v2f b0; b0[0] = Bs[cur][c2][nc];      b0[1] = Bs[cur][c2 + 1][nc];
      v2f b1; b1[0] = Bs[cur][c2][nc + 16]; b1[1] = Bs[cur][c2 + 1][nc + 16];
      acc[0][0] = wmma4(a0, b0, acc[0][0]);
      acc[0][1] = wmma4(a0, b1, acc[0][1]);
      acc[1][0] = wmma4(a1, b0, acc[1][0]);
      acc[1][1] = wmma4(a1, b1, acc[1][1]);
    }
    __syncthreads();  // all waves done reading cur before it is overwritten
  }

#pragma unroll
  for (int tm = 0; tm < 2; tm++)
#pragma unroll
    for (int tn = 0; tn < 2; tn++)
#pragma unroll
      for (int i = 0; i < 8; i++) {
        const int row = mblock + wm + tm * 16 + i + ((lane >> 4) << 3);
        const int col = nblock + wn + tn * 16 + (lane & 15);
        float v = acc[tm][tn][i] * alpha;
        if (Z) v += beta * Z[(long)blockIdx.z * sZ + (long)row * ldz + col];
        Cb[(long)row * ldc + col] = v;
      }
}

// ---------------------------------------------------------------------------
// Attention with the reference's head-mixing grouped softmax:
//   scores s[b,l,m,h] = (q.k)/16 ; softmax over {m in [1024g,1024(g+1)), all h}
//   out[b,l,h,d] = sum_g (1/den_g) * sum_{m in g} exp(s - max_g) * v[b,m,h,d]
// One workgroup per (b, 32 l-rows). 8 waves = 4 heads x 2 halves.
// Online (flash) max/sum per 1024-row group; group acc + final acc in VGPRs.
// Output written into T0 slice of tcat for the second Chebyshev pass.
// ---------------------------------------------------------------------------
__global__ __launch_bounds__(256) void attn_kernel(
    const float* __restrict__ qkv, float* __restrict__ tcat) {
  __shared__ float q_s[H_][32][64];
  __shared__ float k_s[H_][32][64];
  __shared__ float v_s[H_][32][64];
  __shared__ float p_s[H_][32][32];
  __shared__ float red_s[32][8];
  __shared__ float run_max[32], run_sum[32], scale_s[32], inv_s[32];

  const int b   = blockIdx.y;
  const int l0  = blockIdx.x * 32;
  const int tid = threadIdx.x;
  const int lane = tid & 31;
  const int w    = tid >> 5;
  const int hW  = w >> 1;     // wave's head
  const int sub = w & 1;      // m-half for scores / d-half for PV

  // Load Q tile: q_s[h][i][d] = qkv[(b*L+l0+i)*768 + h*192 + d]
#pragma unroll
  for (int r = 0; r < 8; r++) {
    const int idx = tid + r * 256;          // 2048 float4
    const int d4 = idx & 15, i = (idx >> 4) & 31, h = idx >> 9;
    *(float4*)&q_s[h][i][d4 * 4] =
        *(const float4*)&qkv[((long)(b * L_ + l0 + i)) * 768 + h * 192 + d4 * 4];
  }

  v8f gacc[2][2], facc[2][2];
#pragma unroll
  for (int tm = 0; tm < 2; tm++)
#pragma unroll
    for (int tn = 0; tn < 2; tn++) { gacc[tm][tn] = vzero8(); facc[tm][tn] = vzero8(); }

  const int lr = tid >> 3, sl = tid & 7;    // reduction slice: 8 threads per l-row

  for (int g = 0; g < 4; g++) {
    if (tid < 32) { run_max[tid] = -3.0e38f; run_sum[tid] = 0.f; }
    __syncthreads();

    for (int mt = 0; mt < 32; mt++) {
      const int m0 = g * 1024 + mt * 32;
      // Load K,V tiles for all heads
#pragma unroll
      for (int r = 0; r < 8; r++) {
        const int idx = tid + r * 256;
        const int d4 = idx & 15, i = (idx >> 4) & 31, h = idx >> 9;
        const long base = ((long)(b * L_ + m0 + i)) * 768 + h * 192;
        *(float4*)&k_s[h][i][d4 * 4] = *(const float4*)&qkv[base + 64 + d4 * 4];
        *(float4*)&v_s[h][i][d4 * 4] = *(const float4*)&qkv[base + 128 + d4 * 4];
      }
      __syncthreads();

      // Scores: wave (hW, sub) computes s[32 l x 16 m], K-dim = d = 64
      v8f s0 = vzero8(), s1 = vzero8();
#pragma unroll
      for (int kk = 0; kk < 16; kk++) {
        const int c2 = kk * 4 + ((lane >> 4) << 1);
        const int ar = lane & 15;
        v2f a0 = *(v2f*)&q_s[hW][ar][c2];
        v2f a1 = *(v2f*)&q_s[hW][ar + 16][c2];
        const int nc = sub * 16 + (lane & 15);
        v2f bf; bf[0] = k_s[hW][nc][c2]; bf[1] = k_s[hW][nc][c2 + 1];  // K^T
        s0 = wmma4(a0, bf, s0);
        s1 = wmma4(a1, bf, s1);
      }
#pragma unroll
      for (int i = 0; i < 8; i++) {
        const int r0 = i + ((lane >> 4) << 3);
        const int col = sub * 16 + (lane & 15);
        p_s[hW][r0][col]      = s0[i] * 0.0625f;   // 1/sqrt(C)
        p_s[hW][r0 + 16][col] = s1[i] * 0.0625f;
      }
      __syncthreads();

      // Tile max over all heads+m for each l-row
      float mx = -3.0e38f;
      for (int q2 = 0; q2 < 16; q2++) {
        const int j = sl * 16 + q2;             // j in [0,128): h=j>>5, m=j&31
        mx = fmaxf(mx, p_s[j >> 5][lr][j & 31]);
      }
      red_s[lr][sl] = mx;
      __syncthreads();
      if (sl == 0) {
        float tmax = red_s[lr][0];
        for (int q2 = 1; q2 < 8; q2++) tmax = fmaxf(tmax, red_s[lr][q2]);
        const float om = run_max[lr];
        const float nm = fmaxf(om, tmax);
        scale_s[lr] = __expf(om - nm);
        run_max[lr] = nm;
      }
      __syncthreads();

      // Exponentiate in place + tile sum
      float ssum = 0.f;
      const float nm = run_max[lr];
      for (int q2 = 0; q2 < 16; q2++) {
        const int j = sl * 16 + q2;
        const float e = __expf(p_s[j >> 5][lr][j & 31] - nm);
        p_s[j >> 5][lr][j & 31] = e;
        ssum += e;
      }
      red_s[lr][sl] = ssum;
      __syncthreads();
      if (sl == 0) {
        float ts = 0.f;
        for (int q2 = 0; q2 < 8; q2++) ts += red_s[lr][q2];
        run_sum[lr] = run_sum[lr] * scale_s[lr] + ts;
      }
      __syncthreads();

      // Rescale group accumulator, then P@V: wave (hW, sub=d-half)
#pragma unroll
      for (int tm = 0; tm < 2; tm++)
#pragma unroll
        for (int tn = 0; tn < 2; tn++)
#pragma unroll
          for (int i = 0; i < 8; i++) {
            const int row = tm * 16 + i + ((lane >> 4) << 3);
            gacc[tm][tn][i] *= scale_s[row];
          }
#pragma unroll
      for (int kk = 0; kk < 8; kk++) {
        const int c2 = kk * 4 + ((lane >> 4) << 1);
        v2f a0 = *(v2f*)&p_s[hW][lane & 15][c2];
        v2f a1 = *(v2f*)&p_s[hW][(lane & 15) + 16][c2];
        const int dc = sub * 32 + (lane & 15);
        v2f b0; b0[0] = v_s[hW][c2][dc];      b0[1] = v_s[hW][c2 + 1][dc];
        v2f b1; b1[0] = v_s[hW][c2][dc + 16]; b1[1] = v_s[hW][c2 + 1][dc + 16];
        gacc[0][0] = wmma4(a0, b0, gacc[0][0]);
        gacc[0][1] = wmma4(a0, b1, gacc[0][1]);
        gacc[1][0] = wmma4(a1, b0, gacc[1][0]);
        gacc[1][1] = wmma4(a1, b1, gacc[1][1]);
      }
      __syncthreads();
    }

    // Finalize group: divide by group denominator, fold into final acc
    if (tid < 32) inv_s[tid] = 1.f / run_sum[tid];
    __syncthreads();
#pragma unroll
    for (int tm = 0; tm < 2; tm++)
#pragma unroll
      for (int tn = 0; tn < 2; tn++)
#pragma unroll
        for (int i = 0; i < 8; i++) {
          const int row = tm * 16 + i + ((lane >> 4) << 3);
          facc[tm][tn][i] += gacc[tm][tn][i] * inv_s[row];
          gacc[tm][tn][i] = 0.f;
        }
    __syncthreads();
  }

  // Write out into T0 slice of tcat: c = head*64 + d
#pragma unroll
  for (int tm = 0; tm < 2; tm++)
#pragma unroll
    for (int tn = 0; tn < 2; tn++)
#pragma unroll
      for (int i = 0; i < 8; i++) {
        const int row = tm * 16 + i + ((lane >> 4) << 3);
        const int col = hW * 64 + sub * 32 + tn * 16 + (lane & 15);
        tcat[((long)(b * L_ + l0 + row)) * KC_ + col] = facc[tm][tn][i];
      }
}

// ---------------------------------------------------------------------------
// Host-side orchestration
// ---------------------------------------------------------------------------
static void launch_cheb(const float* lap, float* tc, hipStream_t stream) {
  const long sT = (long)L_ * KC_;     // per-batch stride within tcat
  dim3 grid(C_ / 64, L_ / 128, B_);
  // T1 = lap @ T0
  gemm_wmma_kernel<<<grid, 256, 0, stream>>>(
      lap, L_, 0L, tc + 0, KC_, sT, tc + C_, KC_, sT,
      nullptr, 0, 0L, L_, 1.f, 0.f);
  // T2 = 2*lap @ T1 - T0
  gemm_wmma_kernel<<<grid, 256, 0, stream>>>(
      lap, L_, 0L, tc + C_, KC_, sT, tc + 2 * C_, KC_, sT,
      tc + 0, KC_, sT, L_, 2.f, -1.f);
  // T3 = 2*lap @ T2 - T1
  gemm_wmma_kernel<<<grid, 256, 0, stream>>>(
      lap, L_, 0L, tc + 2 * C_, KC_, sT, tc + 3 * C_, KC_, sT,
      tc + C_, KC_, sT, L_, 2.f, -1.f);
}

extern "C" void kernel_launch(void* const* d_in, const int* in_sizes, int n_in,
                              void* d_out, int out_size, void* d_ws, size_t ws_size,
                              hipStream_t stream) {
  const float* x     = (const float*)d_in[0];   // [B,L,C]
  const float* lap   = (const float*)d_in[1];   // [L,L]
  const float* gamma = (const float*)d_in[2];   // [C]
  const float* beta  = (const float*)d_in[3];   // [C]
  const float* wqkv  = (const float*)d_in[4];   // [K,C,3C] -> flat [KC, 3C]
  const float* wout  = (const float*)d_in[5];   // [K,C,C]  -> flat [KC, C]
  float* out = (float*)d_out;                   // [B,L,C]

  // Workspace layout (floats): stats[1024] | tcat[B*L*KC] | qkv[B*L*3C]
  float* stats = (float*)d_ws;
  float* tcat  = stats + 1024;
  float* qkv   = tcat + (long)B_ * L_ * KC_;    // needs ~57 MB total

  // 1) GroupNorm -> T0 slice
  gn_stats_kernel<<<B_ * NG_, 256, 0, stream>>>(x, stats);
  gn_apply_kernel<<<(B_ * L_ * C_) / 256, 256, 0, stream>>>(x, gamma, beta, stats, tcat);

  // 2) Chebyshev basis (pass 1), then qkv = Tcat @ Wqkv
  launch_cheb(lap, tcat, stream);
  gemm_wmma_kernel<<<dim3((3 * C_) / 64, (B_ * L_) / 128, 1), 256, 0, stream>>>(
      tcat, KC_, 0L, wqkv, 3 * C_, 0L, qkv, 3 * C_, 0L,
      nullptr, 0, 0L, KC_, 1.f, 0.f);

  // 3) Attention (grouped softmax), result into T0 slice of tcat
  attn_kernel<<<dim3(L_ / 32, B_), 256, 0, stream>>>(qkv, tcat);

  // 4) Chebyshev basis (pass 2), then out = Tcat @ Wout + x
  launch_cheb(lap, tcat, stream);
  gemm_wmma_kernel<<<dim3(C_ / 64, (B_ * L_) / 128, 1), 256, 0, stream>>>(
      tcat, KC_, 0L, wout, C_, 0L, out, C_, 0L,
      x, C_, 0L, KC_, 1.f, 1.f);
}